// EdgeFeat_42691974922540
// MI455X (gfx1250) — compile-verified
//
#include <hip/hip_runtime.h>
#include <math.h>

typedef __attribute__((ext_vector_type(2))) float v2f;
typedef __attribute__((ext_vector_type(8))) float v8f;

#define N_NODES 20000
#define N_EDGES 320000
#define NBATCH  64
#define NODE_IN 260
#define DDIM    128
#define CONDD   512
#define GEO_OUT 30
#define GEO_K   32   /* K padded to multiple of 4 (and 16) with zeros */

__device__ __forceinline__ v8f wmma_f32(v2f a, v2f b, v8f c) {
  // V_WMMA_F32_16X16X4_F32 : D = A(16x4) * B(4x16) + C(16x16), fp32
  return __builtin_amdgcn_wmma_f32_16x16x4_f32(
      /*neg_a=*/false, a, /*neg_b=*/false, b,
      /*c_mod=*/(short)0, c, /*reuse_a=*/false, /*reuse_b=*/false);
}

// Pair-interleaved transposed weight layout:
//   wP[(c>>1)*(2*W) + 2*j + (c&1)] = w[j][c]   (W = output width)
// so the B operand (w[j][ka], w[j][ka+1]) for even ka is ONE aligned b64 load.

// ---------------- weight prep kernels (tiny) ----------------

// weight-norm node weights into paired-transposed layout (W = 128)
__global__ void prep_nw(const float* __restrict__ n_v, const float* __restrict__ n_g,
                        float* __restrict__ nwP) {
  int j = blockIdx.x;                 // 0..127
  __shared__ float red[64];
  float s = 0.f;
  for (int c = threadIdx.x; c < NODE_IN; c += 64) {
    float v = n_v[j * NODE_IN + c];
    s += v * v;
  }
  red[threadIdx.x] = s;
  __syncthreads();
  for (int off = 32; off > 0; off >>= 1) {
    if ((int)threadIdx.x < off) red[threadIdx.x] += red[threadIdx.x + off];
    __syncthreads();
  }
  float scale = n_g[j] / sqrtf(red[0]);
  for (int c = threadIdx.x; c < NODE_IN; c += 64)
    nwP[(c >> 1) * (2 * DDIM) + 2 * j + (c & 1)] = n_v[j * NODE_IN + c] * scale;
}

// geo weights, paired-transposed (W = 32), rows j>=30 zero-padded
__global__ void prep_geo(const float* __restrict__ geo_v, const float* __restrict__ geo_g,
                         float* __restrict__ geoP) {
  int j = blockIdx.x;                 // 0..31
  int lane = threadIdx.x;             // 0..31
  if (j >= GEO_OUT) {
    if (lane < 8) geoP[(lane >> 1) * (2 * GEO_K) + 2 * j + (lane & 1)] = 0.f;
    return;
  }
  float v = (lane < 8) ? geo_v[j * 8 + lane] : 0.f;
  float s = v * v;
  for (int m = 16; m > 0; m >>= 1) s += __shfl_xor(s, m, 32);
  float scale = geo_g[j] / sqrtf(s);
  if (lane < 8)
    geoP[(lane >> 1) * (2 * GEO_K) + 2 * j + (lane & 1)] = geo_v[j * 8 + lane] * scale;
}

// film_w column blocks into paired-transposed layout (W = 128); w3 K-padded to 32
__global__ void prep_film(const float* __restrict__ film_w, float* __restrict__ w1P,
                          float* __restrict__ w2P, float* __restrict__ w3P) {
  int idx = blockIdx.x * 256 + threadIdx.x;
  if (idx < DDIM * DDIM) {
    int c = idx / DDIM, j = idx % DDIM;
    int o = (c >> 1) * (2 * DDIM) + 2 * j + (c & 1);
    w1P[o] = film_w[j * 286 + c];
    w2P[o] = film_w[j * 286 + 128 + c];
  }
  if (idx < GEO_K * DDIM) {
    int c = idx / DDIM, j = idx % DDIM;
    int o = (c >> 1) * (2 * DDIM) + 2 * j + (c & 1);
    w3P[o] = (c < GEO_OUT) ? film_w[j * 286 + 256 + c] : 0.f;
  }
}

__global__ void prep_cp_scale(const float* __restrict__ cp_v, const float* __restrict__ cp_g,
                              float* __restrict__ scale) {
  int j = blockIdx.x;                 // 0..255
  __shared__ float red[128];
  float s = 0.f;
  for (int c = threadIdx.x; c < CONDD; c += 128) {
    float v = cp_v[j * CONDD + c];
    s += v * v;
  }
  red[threadIdx.x] = s;
  __syncthreads();
  for (int off = 64; off > 0; off >>= 1) {
    if ((int)threadIdx.x < off) red[threadIdx.x] += red[threadIdx.x + off];
    __syncthreads();
  }
  if (threadIdx.x == 0) scale[j] = cp_g[j] / sqrtf(red[0]);
}

// gb[b, j] = dot(cond[b], cp_v[j]) * scale[j] + cp_b[j]   (64 x 256)
__global__ void cond_gb(const float* __restrict__ cond, const float* __restrict__ cp_v,
                        const float* __restrict__ cp_b, const float* __restrict__ scale,
                        float* __restrict__ gb) {
  int j = blockIdx.x;                 // 0..255
  int b = blockIdx.y;                 // 0..63
  __shared__ float red[128];
  float s = 0.f;
  for (int c = threadIdx.x; c < CONDD; c += 128)
    s += cond[b * CONDD + c] * cp_v[j * CONDD + c];
  red[threadIdx.x] = s;
  __syncthreads();
  for (int off = 64; off > 0; off >>= 1) {
    if ((int)threadIdx.x < off) red[threadIdx.x] += red[threadIdx.x + off];
    __syncthreads();
  }
  if (threadIdx.x == 0) gb[b * 256 + j] = red[0] * scale[j] + cp_b[j];
}

// ---------------- node pipeline: h = relu(x@nw^T + b); A = h@W1^T; B = h@W2^T ----------------

__global__ void __launch_bounds__(32) node_fwd(const float* __restrict__ x,
                                               const float* __restrict__ n_b,
                                               const float* __restrict__ nwP,
                                               const float* __restrict__ w1P,
                                               const float* __restrict__ w2P,
                                               float* __restrict__ Aout,
                                               float* __restrict__ Bout) {
  __shared__ float hs[16 * DDIM];     // h tile, row-major 16x128
  const int lane  = threadIdx.x;
  const int lrow  = lane & 15;
  const int lhalf = lane >> 4;
  const int base  = blockIdx.x * 16;

  const float* xrow = x + (size_t)(base + lrow) * NODE_IN;

  // h tile (16 x 128): 8 column tiles, K = 260 in steps of 4
  for (int nt = 0; nt < 8; ++nt) {
    v8f c = {0.f, 0.f, 0.f, 0.f, 0.f, 0.f, 0.f, 0.f};
    const int jn = nt * 16 + lrow;
    for (int k0 = 0; k0 < NODE_IN; k0 += 4) {
      const int ka = k0 + lhalf * 2;        // even
      v2f a = *(const v2f*)(xrow + ka);
      v2f b = *(const v2f*)(nwP + (size_t)(ka >> 1) * (2 * DDIM) + 2 * jn);
      c = wmma_f32(a, b, c);
    }
    const float bias = n_b[jn];
#pragma unroll
    for (int r = 0; r < 8; ++r) {
      float v = c[r] + bias;
      hs[(r + lhalf * 8) * DDIM + jn] = v > 0.f ? v : 0.f;
    }
  }
  __syncthreads();

  // A = h @ W1^T ; B = h @ W2^T   (16x128 each), K = 128
  for (int nt = 0; nt < 8; ++nt) {
    v8f ca = {0.f, 0.f, 0.f, 0.f, 0.f, 0.f, 0.f, 0.f};
    v8f cb = {0.f, 0.f, 0.f, 0.f, 0.f, 0.f, 0.f, 0.f};
    const int jn = nt * 16 + lrow;
    for (int k0 = 0; k0 < DDIM; k0 += 4) {
      const int ka = k0 + lhalf * 2;        // even
      v2f a  = *(const v2f*)(hs + lrow * DDIM + ka);
      v2f b1 = *(const v2f*)(w1P + (size_t)(ka >> 1) * (2 * DDIM) + 2 * jn);
      v2f b2 = *(const v2f*)(w2P + (size_t)(ka >> 1) * (2 * DDIM) + 2 * jn);
      ca = wmma_f32(a, b1, ca);
      cb = wmma_f32(a, b2, cb);
    }
#pragma unroll
    for (int r = 0; r < 8; ++r) {
      const size_t row = (size_t)(base + r + lhalf * 8);
      Aout[row * DDIM + jn] = ca[r];
      Bout[row * DDIM + jn] = cb[r];
    }
  }
}

// ---------------- edge pipeline: geo GEMM, y = A[src]+B[dst]+geo_h@W3^T+b, LN, FiLM, relu ----------

__global__ void __launch_bounds__(32) edge_fwd(const float* __restrict__ edge_geo,
                                               const int* __restrict__ esrc,
                                               const int* __restrict__ edst,
                                               const int* __restrict__ ebatch,
                                               const float* __restrict__ geo_b,
                                               const float* __restrict__ film_b,
                                               const float* __restrict__ geoP,
                                               const float* __restrict__ w3P,
                                               const float* __restrict__ gb,
                                               const float* __restrict__ Aout,
                                               const float* __restrict__ Bout,
                                               float* __restrict__ out) {
  __shared__ float gs[16 * GEO_K];    // relu(geo linear), padded cols 30,31 = 0
  const int lane  = threadIdx.x;
  const int lrow  = lane & 15;
  const int lhalf = lane >> 4;
  const int base  = blockIdx.x * 16;

  // geo_h = relu(edge_geo @ geo_w^T + geo_b) : (16 x 30), K = 8
  const float* grow = edge_geo + (size_t)(base + lrow) * 8;
  for (int nt = 0; nt < 2; ++nt) {
    v8f c = {0.f, 0.f, 0.f, 0.f, 0.f, 0.f, 0.f, 0.f};
    const int jn = nt * 16 + lrow;
    for (int k0 = 0; k0 < 8; k0 += 4) {
      const int ka = k0 + lhalf * 2;        // even
      v2f a = *(const v2f*)(grow + ka);
      v2f b = *(const v2f*)(geoP + (ka >> 1) * (2 * GEO_K) + 2 * jn);
      c = wmma_f32(a, b, c);
    }
    const bool  valid = (jn < GEO_OUT);
    const float bias  = valid ? geo_b[jn] : 0.f;
#pragma unroll
    for (int r = 0; r < 8; ++r) {
      float v = valid ? (c[r] + bias) : 0.f;
      gs[(r + lhalf * 8) * GEO_K + jn] = v > 0.f ? v : 0.f;
    }
  }
  __syncthreads();

  // y = A[src] + B[dst] + geo_h @ W3^T + film_b : (16 x 128)
  v8f y[8];
  for (int nt = 0; nt < 8; ++nt) {
    v8f c = {0.f, 0.f, 0.f, 0.f, 0.f, 0.f, 0.f, 0.f};
    const int jn = nt * 16 + lrow;
    for (int k0 = 0; k0 < GEO_K; k0 += 4) {
      const int ka = k0 + lhalf * 2;        // even
      v2f a = *(const v2f*)(gs + lrow * GEO_K + ka);
      v2f b = *(const v2f*)(w3P + (ka >> 1) * (2 * DDIM) + 2 * jn);
      c = wmma_f32(a, b, c);
    }
    const float fb = film_b[jn];
#pragma unroll
    for (int r = 0; r < 8; ++r) {
      const int e = base + r + lhalf * 8;
      const int s = esrc[e];
      const int d = edst[e];
      c[r] += Aout[(size_t)s * DDIM + jn] + Bout[(size_t)d * DDIM + jn] + fb;
    }
    y[nt] = c;
  }

  // LayerNorm over 128 per edge row; row values live in (r, lane-half) x 8 tiles
  float mu[8], rstd[8];
#pragma unroll
  for (int r = 0; r < 8; ++r) {
    float s = 0.f;
#pragma unroll
    for (int nt = 0; nt < 8; ++nt) s += y[nt][r];
    s += __shfl_xor(s, 1, 16);
    s += __shfl_xor(s, 2, 16);
    s += __shfl_xor(s, 4, 16);
    s += __shfl_xor(s, 8, 16);
    mu[r] = s * (1.0f / 128.0f);
    float v = 0.f;
#pragma unroll
    for (int nt = 0; nt < 8; ++nt) {
      float d = y[nt][r] - mu[r];
      v += d * d;
    }
    v += __shfl_xor(v, 1, 16);
    v += __shfl_xor(v, 2, 16);
    v += __shfl_xor(v, 4, 16);
    v += __shfl_xor(v, 8, 16);
    rstd[r] = rsqrtf(v * (1.0f / 128.0f) + 1e-5f);
  }

  // FiLM + relu + store
  for (int nt = 0; nt < 8; ++nt) {
    const int jn = nt * 16 + lrow;
#pragma unroll
    for (int r = 0; r < 8; ++r) {
      const int e = base + r + lhalf * 8;
      const int bi = ebatch[e];
      const float gamma = gb[bi * 256 + jn] + 1.0f;
      const float beta  = gb[bi * 256 + 128 + jn];
      float v = (y[nt][r] - mu[r]) * rstd[r];
      v = gamma * v + beta;
      out[(size_t)e * DDIM + jn] = v > 0.f ? v : 0.f;
    }
  }
}

// ---------------- launch ----------------

extern "C" void kernel_launch(void* const* d_in, const int* in_sizes, int n_in,
                              void* d_out, int out_size, void* d_ws, size_t ws_size,
                              hipStream_t stream) {
  const float* node_feats = (const float*)d_in[0];
  const float* edge_geo   = (const float*)d_in[1];
  const float* cond       = (const float*)d_in[2];
  const int*   esrc       = (const int*)d_in[3];
  const int*   edst       = (const int*)d_in[4];
  const int*   ebatch     = (const int*)d_in[5];
  const float* n_v        = (const float*)d_in[6];
  const float* n_g        = (const float*)d_in[7];
  const float* n_b        = (const float*)d_in[8];
  const float* geo_v      = (const float*)d_in[9];
  const float* geo_g      = (const float*)d_in[10];
  const float* geo_b      = (const float*)d_in[11];
  const float* cp_v       = (const float*)d_in[12];
  const float* cp_g       = (const float*)d_in[13];
  const float* cp_b       = (const float*)d_in[14];
  const float* film_w     = (const float*)d_in[15];
  const float* film_b     = (const float*)d_in[16];

  float* ws      = (float*)d_ws;
  float* nwP     = ws;                               // 260*128 = 33280
  float* geoP    = nwP + NODE_IN * DDIM;             // 8*32    = 256
  float* w1P     = geoP + 8 * GEO_K;                 // 128*128 = 16384
  float* w2P     = w1P + DDIM * DDIM;                // 16384
  float* w3P     = w2P + DDIM * DDIM;                // 32*128  = 4096
  float* cpscale = w3P + GEO_K * DDIM;               // 256
  float* gbbuf   = cpscale + 256;                    // 64*256  = 16384
  float* Aout    = gbbuf + NBATCH * 256;             // N*128
  float* Bout    = Aout + (size_t)N_NODES * DDIM;    // N*128

  prep_nw<<<DDIM, 64, 0, stream>>>(n_v, n_g, nwP);
  prep_geo<<<GEO_K, 32, 0, stream>>>(geo_v, geo_g, geoP);
  prep_film<<<(DDIM * DDIM + 255) / 256, 256, 0, stream>>>(film_w, w1P, w2P, w3P);
  prep_cp_scale<<<256, 128, 0, stream>>>(cp_v, cp_g, cpscale);
  cond_gb<<<dim3(256, NBATCH), 128, 0, stream>>>(cond, cp_v, cp_b, cpscale, gbbuf);

  node_fwd<<<N_NODES / 16, 32, 0, stream>>>(node_feats, n_b, nwP, w1P, w2P, Aout, Bout);
  edge_fwd<<<N_EDGES / 16, 32, 0, stream>>>(edge_geo, esrc, edst, ebatch, geo_b, film_b,
                                            geoP, w3P, gbbuf, Aout, Bout, (float*)d_out);
}